// DynamicsWithKnownReward_14542759264837
// MI455X (gfx1250) — compile-verified
//
#include <hip/hip_runtime.h>
#include <hip/hip_bf16.h>
#include <math.h>

typedef __attribute__((ext_vector_type(2))) float v2f;
typedef __attribute__((ext_vector_type(8))) float v8f;

#define HID 128
#define WAVES 8
#define TPB 256
#define ROWS_PER_BLOCK 1024
#define ITERS (ROWS_PER_BLOCK / (WAVES * 16))   // 8
#define HSTRIDE 132                             // 128 + 4 pad: kills LDS bank conflicts

static __device__ __forceinline__ v8f wmma_f32(v2f a, v2f b, v8f c) {
    // V_WMMA_F32_16X16X4_F32 : D(16x16,f32) = A(16x4,f32) * B(4x16,f32) + C
    return __builtin_amdgcn_wmma_f32_16x16x4_f32(
        /*neg_a=*/false, a, /*neg_b=*/false, b,
        /*c_mod=*/(short)0, c, /*reuse_a=*/false, /*reuse_b=*/false);
}

__global__ __launch_bounds__(TPB)
void dyn_mlp_reward(const float* __restrict__ S,
                    const float* __restrict__ A,
                    const float* __restrict__ W1,
                    const float* __restrict__ B1,
                    const float* __restrict__ W2,
                    const float* __restrict__ B2,
                    const float* __restrict__ W3,
                    const float* __restrict__ B3,
                    float* __restrict__ OutS,
                    float* __restrict__ OutR,
                    int nrows)
{
    // Weights in WMMA B-fragment layout:
    //   fragment(kb4, n)[lane] = (W[k][col], W[k+1][col]),
    //   k = 4*kb4 + 2*(lane>=16), col = 16*n + (lane&15)
    __shared__ v2f  W1s[2 * 8 * 32];     //  4 KB (K padded 5 -> 8)
    __shared__ v2f  W2s[32 * 8 * 32];    // 64 KB
    __shared__ v2f  W3s[32 * 32];        //  8 KB (N padded 3 -> 16)
    __shared__ float hbuf[WAVES][16 * HSTRIDE]; // per-wave activation transpose buffer
    __shared__ float b1s[HID], b2s[HID], b3s[16];

    const int tid = threadIdx.x;
    const int blockBase = blockIdx.x * ROWS_PER_BLOCK;

    // ------------------- reward (elementwise, all 256 threads) -------------------
    {
        const float ox[6] = {0.f, 0.f, 0.f, 0.f, 0.f, -0.8f};
        const float oy[6] = {0.f, 0.2f, 0.4f, 0.6f, 0.8f, -0.8f};
        const float inv2var = 0.5f / 0.035f;
        const float gcoef   = 100.0f / (2.0f * 3.14159265358979323f * 0.035f);
        const float bcoef   = 10.0f / (0.03f * 2.5066282746310002f); // 10/(SIG*sqrt(2pi))
        const float ibs     = 0.5f / (0.03f * 0.03f);
#pragma unroll
        for (int i = 0; i < ROWS_PER_BLOCK / TPB; ++i) {
            int r = blockBase + i * TPB + tid;
            if (r < nrows) {
                float sx = S[r*3+0], sy = S[r*3+1];
                float ax = A[r*2+0], ay = A[r*2+1];
                float gs = 0.f;
#pragma unroll
                for (int o = 0; o < 6; ++o) {
                    float dx = sx - ox[o], dy = sy - oy[o];
                    gs += __expf(-(dx*dx + dy*dy) * inv2var);
                }
                float quad = 30.f * ((sx-ax)*(sx-ax) + (sy-ay)*(sy-ay));
                float e0 = __expf(-(sx+1.5f)*(sx+1.5f) * ibs);
                float e1 = __expf(-(sx-1.5f)*(sx-1.5f) * ibs);
                float e2 = __expf(-(sy-1.0f)*(sy-1.0f) * ibs);
                float e3 = __expf(-(sy+1.0f)*(sy+1.0f) * ibs);
                OutR[r] = -(quad + gcoef * gs + bcoef * (e0+e1+e2+e3));
            }
        }
    }

    // ------------------- stage weights into LDS (B-fragment layout) --------------
    for (int e = tid; e < 32*8*32; e += TPB) {
        int kb4 = e >> 8, n = (e >> 5) & 7, l = e & 31;
        int k   = kb4*4 + ((l >> 4) << 1);
        int col = n*16 + (l & 15);
        v2f v; v.x = W2[k*HID + col]; v.y = W2[(k+1)*HID + col];
        W2s[e] = v;
    }
    for (int e = tid; e < 2*8*32; e += TPB) {
        int kb4 = e >> 8, n = (e >> 5) & 7, l = e & 31;
        int k   = kb4*4 + ((l >> 4) << 1);
        int col = n*16 + (l & 15);
        v2f v;
        v.x = (k   < 5) ? W1[k*HID + col]     : 0.f;
        v.y = (k+1 < 5) ? W1[(k+1)*HID + col] : 0.f;
        W1s[e] = v;
    }
    for (int e = tid; e < 32*32; e += TPB) {
        int kb4 = e >> 5, l = e & 31;
        int k   = kb4*4 + ((l >> 4) << 1);
        int col = l & 15;
        v2f v;
        v.x = (col < 3) ? W3[k*3 + col]     : 0.f;
        v.y = (col < 3) ? W3[(k+1)*3 + col] : 0.f;
        W3s[e] = v;
    }
    if (tid < HID) { b1s[tid] = B1[tid]; b2s[tid] = B2[tid]; }
    if (tid < 16)  { b3s[tid] = (tid < 3) ? B3[tid] : 0.f; }
    __syncthreads();

    const int w  = tid >> 5;
    const int l  = tid & 31;
    const int hi = l >> 4;        // 0: lanes 0-15, 1: lanes 16-31
    const int lm = l & 15;
    float* hb = hbuf[w];

    for (int it = 0; it < ITERS; ++it) {
        const int rowbase = blockBase + (w * ITERS + it) * 16;
        if (rowbase + 16 > nrows) break;    // wave-uniform

        // ---------------- layer 1: x(16x5 pad 8) @ W1 -> h1(16x128) ----------------
        // A fragment layout: lane<16 holds (M=lane, K=kb..kb+1), lane>=16 (K=kb+2..kb+3)
        const int r0 = rowbase + lm;        // x = [s0 s1 s2 a0 a1 | 0 0 0]
        float x0 = hi ? S[r0*3+2] : S[r0*3+0];
        float x1 = hi ? A[r0*2+0] : S[r0*3+1];
        float x2 = hi ? 0.f       : A[r0*2+1];
        v2f af0; af0.x = x0; af0.y = x1;    // K block 0..3
        v2f af1; af1.x = x2; af1.y = 0.f;   // K block 4..7 (padded)

#pragma unroll
        for (int n = 0; n < 8; ++n) {
            v8f c = {0.f,0.f,0.f,0.f,0.f,0.f,0.f,0.f};
            c = wmma_f32(af0, W1s[(0*8 + n)*32 + l], c);
            c = wmma_f32(af1, W1s[(1*8 + n)*32 + l], c);
            float bv = b1s[n*16 + lm];
            // D layout: VGPR g, lane<16 -> (M=g, N=lane); lane>=16 -> (M=8+g, N=lane-16)
#pragma unroll
            for (int g = 0; g < 8; ++g)
                hb[(g + 8*hi)*HSTRIDE + n*16 + lm] = fmaxf(c[g] + bv, 0.f);
        }

        // ---------------- layer 2: h1 @ W2 -> h2(16x128), k-outer ------------------
        v8f c2[8];
#pragma unroll
        for (int n = 0; n < 8; ++n)
            c2[n] = (v8f){0.f,0.f,0.f,0.f,0.f,0.f,0.f,0.f};
        for (int kb4 = 0; kb4 < 32; ++kb4) {
            v2f af = *(const v2f*)&hb[lm*HSTRIDE + kb4*4 + 2*hi]; // one ds_load_b64
#pragma unroll
            for (int n = 0; n < 8; ++n)
                c2[n] = wmma_f32(af, W2s[(kb4*8 + n)*32 + l], c2[n]);
        }
        // bias + relu, overwrite hb with h2 (same-wave LDS ops are in-order)
#pragma unroll
        for (int n = 0; n < 8; ++n) {
            float bv = b2s[n*16 + lm];
#pragma unroll
            for (int g = 0; g < 8; ++g)
                hb[(g + 8*hi)*HSTRIDE + n*16 + lm] = fmaxf(c2[n][g] + bv, 0.f);
        }

        // ---------------- layer 3: h2 @ W3 -> s_next(16x3, N padded 16) ------------
        v8f c3 = {0.f,0.f,0.f,0.f,0.f,0.f,0.f,0.f};
        for (int kb4 = 0; kb4 < 32; ++kb4) {
            v2f af = *(const v2f*)&hb[lm*HSTRIDE + kb4*4 + 2*hi];
            c3 = wmma_f32(af, W3s[kb4*32 + l], c3);
        }
        if (lm < 3) {
            float bv = b3s[lm];
#pragma unroll
            for (int g = 0; g < 8; ++g)
                OutS[(rowbase + g + 8*hi)*3 + lm] = c3[g] + bv;
        }
    }
}

extern "C" void kernel_launch(void* const* d_in, const int* in_sizes, int n_in,
                              void* d_out, int out_size, void* d_ws, size_t ws_size,
                              hipStream_t stream)
{
    const float* S  = (const float*)d_in[0];
    const float* A  = (const float*)d_in[1];
    const float* W1 = (const float*)d_in[2];
    const float* B1 = (const float*)d_in[3];
    const float* W2 = (const float*)d_in[4];
    const float* B2 = (const float*)d_in[5];
    const float* W3 = (const float*)d_in[6];
    const float* B3 = (const float*)d_in[7];
    const int nrows = in_sizes[0] / 3;           // B = 524288
    float* outS = (float*)d_out;                 // (B,3)
    float* outR = (float*)d_out + (size_t)nrows * 3; // (B,1)
    const int grid = (nrows + ROWS_PER_BLOCK - 1) / ROWS_PER_BLOCK;
    hipLaunchKernelGGL(dyn_mlp_reward, dim3(grid), dim3(TPB), 0, stream,
                       S, A, W1, B1, W2, B2, W3, B3, outS, outR, nrows);
}